// Rasterize_51908974739476
// MI455X (gfx1250) — compile-verified
//
#include <hip/hip_runtime.h>
#include <hip/hip_bf16.h>
#include <stdint.h>

// Problem constants (match reference)
#define BB   2
#define FF   512
#define HH   192
#define WW   192
#define TT   8
#define FARV 100.0f
#define EPSV 1e-8f

typedef __attribute__((ext_vector_type(2))) float v2f;
typedef __attribute__((ext_vector_type(8))) float v8f;
typedef __attribute__((ext_vector_type(4))) unsigned int u32x4;
typedef __attribute__((ext_vector_type(8))) int i32x8;
typedef __attribute__((ext_vector_type(4))) int i32x4;

#define STRIPS_PER_ROW (WW / 16)                    // 12
#define STRIPS_TOTAL   (BB * HH * STRIPS_PER_ROW)   // 4608
#define WAVES_PER_BLK  8
#define FWORDS         (FF * 12)                    // 6144 dwords of face data per batch

#ifndef __has_builtin
#define __has_builtin(x) 0
#endif

#if __has_builtin(__builtin_amdgcn_tensor_load_to_lds) && \
    __has_builtin(__builtin_amdgcn_s_wait_tensorcnt)
#define HAVE_TDM 1
#else
#define HAVE_TDM 0
#warning "CDNA5 probe: __builtin_amdgcn_tensor_load_to_lds NOT available; using ds-store staging fallback"
#endif

// ---------------------------------------------------------------------------
// Kernel 1: per-face preprocessing.
// w0(p,f) = c0 . [1,px,py]; w1(p,f) = c1 . [1,px,py]; invd(p,f) = d . [1,px,py]
// Degenerate faces get c0 = {-1,0,0} so the inside test always fails.
// Layout: fdata[(b*FF+f)*12 + {0..2:c0, 3..5:c1, 6..8:d, 9..11:pad}]
// ---------------------------------------------------------------------------
__global__ void prep_kernel(const float* __restrict__ faces,
                            float* __restrict__ fdata) {
    int i = blockIdx.x * blockDim.x + threadIdx.x;   // b*FF + f
    if (i >= BB * FF) return;
    const float* v = faces + (size_t)i * 9;
    float x0 = v[0], y0 = v[1], z0 = v[2];
    float x1 = v[3], y1 = v[4], z1 = v[5];
    float x2 = v[6], y2 = v[7], z2 = v[8];

    float det = (x1 - x0) * (y2 - y0) - (x2 - x0) * (y1 - y0);
    bool  ok  = fabsf(det) > EPSV;
    float inv_det = ok ? (1.0f / det) : 1.0f;

    float c00 = (x1 * y2 - x2 * y1) * inv_det;
    float c01 = (y1 - y2) * inv_det;
    float c02 = (x2 - x1) * inv_det;
    float c10 = (x2 * y0 - x0 * y2) * inv_det;
    float c11 = (y2 - y0) * inv_det;
    float c12 = (x0 - x2) * inv_det;

    float rz0 = 1.0f / z0, rz1 = 1.0f / z1, rz2 = 1.0f / z2;
    float ka = rz0 - rz2, kb = rz1 - rz2;
    float d0 = c00 * ka + c10 * kb + rz2;
    float d1 = c01 * ka + c11 * kb;
    float d2 = c02 * ka + c12 * kb;

    float* o = fdata + (size_t)i * 12;
    if (ok) {
        o[0] = c00; o[1] = c01; o[2] = c02;
        o[3] = c10; o[4] = c11; o[5] = c12;
        o[6] = d0;  o[7] = d1;  o[8] = d2;
    } else {
        o[0] = -1.0f; o[1] = 0.0f; o[2] = 0.0f;
        o[3] = 0.0f;  o[4] = 0.0f; o[5] = 0.0f;
        o[6] = 0.0f;  o[7] = 0.0f; o[8] = 0.0f;
    }
    o[9] = 0.0f; o[10] = 0.0f; o[11] = 0.0f;
}

// ---------------------------------------------------------------------------
// Kernel 2: WMMA rasterizer. One wave = one 16-pixel strip (M), loop over 32
// tiles of 16 faces (N). Face-coefficient table staged into LDS (TDM when the
// builtin exists, ds-store loop otherwise); three v_wmma_f32_16x16x4_f32 per
// tile produce w0 / w1 / inv_depth for each 16x16 (pixel,face) tile; VALU
// does a 2x min3 validity gate and packed-u64 argmin (max invd, tie -> lowest
// face index).
// ---------------------------------------------------------------------------
__global__ __launch_bounds__(256) void raster_kernel(
        const float* __restrict__ fdata,
        float* __restrict__ out_fim,    // face index as float, B*H*W
        float* __restrict__ out_depth,  // depth map, B*H*W
        float* __restrict__ ws_w0,      // best-face w0 per pixel
        float* __restrict__ ws_w1) {    // best-face w1 per pixel
    __shared__ float lds[FWORDS];       // 24 KB face-coefficient table

    const int lane = threadIdx.x & 31;
    const int wave = threadIdx.x >> 5;
    const int strip = blockIdx.x * WAVES_PER_BLK + wave;
    const int b   = strip / (HH * STRIPS_PER_ROW);   // uniform per block (288 blocks/batch)
    const int rem = strip % (HH * STRIPS_PER_ROW);
    const int row = rem / STRIPS_PER_ROW;
    const int xbase = (rem % STRIPS_PER_ROW) * 16;

    const float* src = fdata + (size_t)b * FWORDS;

#if HAVE_TDM
    // --- Stage face table into LDS with the Tensor Data Mover -------------
    if (wave == 0) {
        unsigned long long ga = (unsigned long long)(uintptr_t)src;
        // Flat LDS addresses truncate to the LDS byte offset in [31:0].
        unsigned int lds_base = (unsigned int)(uintptr_t)(void*)lds;

        u32x4 g0;
        g0.x = 1u;                                        // count=1, user descriptor
        g0.y = lds_base;                                  // lds_addr
        g0.z = (unsigned int)(ga & 0xFFFFFFFFull);        // global_addr[31:0]
        g0.w = (unsigned int)((ga >> 32) & 0x01FFFFFFull) // global_addr[56:32]
             | (2u << 30);                                // type = 2 ("image")

        i32x8 g1;
        g1[0] = (int)(2u << 16);                          // data_size = 4B
        g1[1] = (int)(((unsigned)FWORDS & 0xFFFFu) << 16);      // tensor_dim0[15:0]
        g1[2] = (int)((((unsigned)FWORDS >> 16) & 0xFFFFu)      // tensor_dim0[31:16]
             | (1u << 16));                               // tensor_dim1 = 1
        g1[3] = (int)(((unsigned)FWORDS & 0xFFFFu) << 16);      // tile_dim0 = 6144
        g1[4] = 1;                                        // tile_dim1 = 1, tile_dim2 = 0
        g1[5] = (int)FWORDS;                              // tensor_dim0_stride[31:0]
        g1[6] = 0;                                        // stride[47:32], dim1_stride[15:0]
        g1[7] = 0;

        i32x4 gz = {0, 0, 0, 0};
#if defined(__clang_major__) && (__clang_major__ >= 23)
        i32x8 gz8 = {0, 0, 0, 0, 0, 0, 0, 0};
        __builtin_amdgcn_tensor_load_to_lds(g0, g1, gz, gz, gz8, 0);
#else
        __builtin_amdgcn_tensor_load_to_lds(g0, g1, gz, gz, 0);
#endif
        __builtin_amdgcn_s_wait_tensorcnt(0);
    }
#else
    for (int i = threadIdx.x; i < FWORDS; i += blockDim.x) lds[i] = src[i];
#endif
    __syncthreads();

    const float py = (2.0f * ((float)row + 0.5f) - (float)HH) / (float)HH;
    const int   m  = lane & 15;
    const float pxm = (2.0f * ((float)(xbase + m) + 0.5f) - (float)WW) / (float)WW;

    // A fragment (16x4 f32): lanes 0-15 hold (K0,K1)=(1,px[M]); lanes 16-31 hold (K2,K3)=(py,0)
    v2f afrag;
    if (lane < 16) { afrag.x = 1.0f; afrag.y = pxm; }
    else           { afrag.x = py;   afrag.y = 0.0f; }

    unsigned long long key[8];
#pragma unroll
    for (int r = 0; r < 8; ++r) key[r] = 0ull;

    const v8f czero = {};

#pragma unroll 2
    for (int t = 0; t < FF / 16; ++t) {
        const int fcol = t * 16 + m;               // face handled by this lane's column
        const float* fb = &lds[fcol * 12];

        // B fragments (4x16 f32): VGPR0 = row (lane<16?0:2), VGPR1 = row (lane<16?1:3)
        v2f b0, b1, b2;
        if (lane < 16) {
            b0.x = fb[0]; b0.y = fb[1];
            b1.x = fb[3]; b1.y = fb[4];
            b2.x = fb[6]; b2.y = fb[7];
        } else {
            b0.x = fb[2]; b0.y = 0.0f;
            b1.x = fb[5]; b1.y = 0.0f;
            b2.x = fb[8]; b2.y = 0.0f;
        }

        v8f w0v = __builtin_amdgcn_wmma_f32_16x16x4_f32(false, afrag, false, b0,
                                                        (short)0, czero, false, false);
        v8f w1v = __builtin_amdgcn_wmma_f32_16x16x4_f32(false, afrag, false, b1,
                                                        (short)0, czero, false, false);
        v8f dv  = __builtin_amdgcn_wmma_f32_16x16x4_f32(false, afrag, false, b2,
                                                        (short)0, czero, false, false);

        const unsigned int fkey = 0xFFFFFFFFu - (unsigned int)fcol;
#pragma unroll
        for (int r = 0; r < 8; ++r) {
            const float w0 = w0v[r], w1 = w1v[r], invd = dv[r];
            const float w2 = 1.0f - w0 - w1;
            // Validity as a single sign test:
            //   inside  <=> min3(w0,w1,w2) >= 0
            //   depth in [NEAR,FAR] <=> invd in [0.01, 2.0]
            //             <=> min(invd-0.01, 2.0-invd) >= 0
            const float mn   = fminf(fminf(w0, w1), w2);
            const float gate = fminf(fminf(mn, invd - 0.01f), 2.0f - invd);
            // Gate only invd: invalid -> high word 0, can never beat a valid key
            // (valid high word >= bits(0.01)), and fails depth<FAR at the end.
            const float invd_ok = (gate >= 0.0f) ? invd : 0.0f;
            const unsigned long long k =
                (((unsigned long long)__float_as_uint(invd_ok)) << 32) | fkey;
            key[r] = (k > key[r]) ? k : key[r];
        }
    }

    // Reduce over the 16 face-columns (lanes of each half-wave), per pixel row.
#pragma unroll
    for (int r = 0; r < 8; ++r) {
        unsigned long long k = key[r];
#pragma unroll
        for (int s = 1; s < 16; s <<= 1) {
            unsigned long long o = __shfl_xor(k, s, 16);
            k = (o > k) ? o : k;
        }
        key[r] = k;
    }

    // Lane 0 finalizes pixels 0..7; lane 16 finalizes pixels 8..15.
    if (m == 0) {
        const int mhalf = (lane >> 4) * 8;
        for (int r = 0; r < 8; ++r) {
            const int xpix = xbase + mhalf + r;
            const int pixl = (b * HH + row) * WW + xpix;
            const unsigned long long k = key[r];
            const float invd = __uint_as_float((unsigned int)(k >> 32));
            const float depth = 1.0f / invd;          // inf when invd==0
            const bool covered = (depth < FARV);
            int fidx = -1;
            float w0 = 0.0f, w1 = 0.0f;
            if (covered) {
                fidx = (int)(0xFFFFFFFFu - (unsigned int)(k & 0xFFFFFFFFu));
                const float* fb = &lds[fidx * 12];
                const float pxr = (2.0f * ((float)xpix + 0.5f) - (float)WW) / (float)WW;
                w0 = fb[0] + fb[1] * pxr + fb[2] * py;
                w1 = fb[3] + fb[4] * pxr + fb[5] * py;
            }
            out_fim[pixl]   = (float)fidx;
            out_depth[pixl] = covered ? depth : FARV;
            ws_w0[pixl] = w0;
            ws_w1[pixl] = w1;
        }
    }
}

// ---------------------------------------------------------------------------
// Kernel 3: texture gather + blend into (B, T+1, H, W) feature planes.
// ---------------------------------------------------------------------------
__global__ void finalize_kernel(const float* __restrict__ textures,
                                const float* __restrict__ fimf,
                                const float* __restrict__ ws_w0,
                                const float* __restrict__ ws_w1,
                                float* __restrict__ feat_out) {
    int p = blockIdx.x * blockDim.x + threadIdx.x;   // over B*H*W
    if (p >= BB * HH * WW) return;
    const int b = p / (HH * WW);
    const int pix = p % (HH * WW);

    const int f = (int)fimf[p];
    const bool covered = (f >= 0);
    const float w0 = ws_w0[p];
    const float w1 = ws_w1[p];
    const float w2 = covered ? (1.0f - w0 - w1) : 0.0f;
    const int fi = covered ? f : 0;

    const float* tex = textures + ((size_t)(b * FF + fi)) * 3 * TT;
    __builtin_prefetch(tex, 0, 0);   // global_prefetch_b8 on the gathered row
#pragma unroll
    for (int t = 0; t < TT; ++t) {
        const float v = w0 * tex[t] + w1 * tex[TT + t] + w2 * tex[2 * TT + t];
        feat_out[((size_t)(b * (TT + 1) + t)) * HH * WW + pix] = v;
    }
    feat_out[((size_t)(b * (TT + 1) + TT)) * HH * WW + pix] = covered ? 1.0f : 0.0f;
}

// ---------------------------------------------------------------------------
// Launcher. d_out layout (floats): [feature B*9*H*W][fim B*H*W][depth B*H*W]
// d_ws layout (floats): [fdata 2*512*12][w0 B*H*W][w1 B*H*W]
// ---------------------------------------------------------------------------
extern "C" void kernel_launch(void* const* d_in, const int* in_sizes, int n_in,
                              void* d_out, int out_size, void* d_ws, size_t ws_size,
                              hipStream_t stream) {
    const float* faces    = (const float*)d_in[0];
    const float* textures = (const float*)d_in[1];
    float* out = (float*)d_out;
    float* ws  = (float*)d_ws;

    float* fdata  = ws;                               // 12288 floats
    float* ws_w0  = ws + (size_t)BB * FWORDS;         // 73728 floats
    float* ws_w1  = ws_w0 + (size_t)BB * HH * WW;     // 73728 floats

    float* out_feat  = out;
    float* out_fim   = out + (size_t)BB * (TT + 1) * HH * WW;
    float* out_depth = out_fim + (size_t)BB * HH * WW;

    prep_kernel<<<(BB * FF + 255) / 256, 256, 0, stream>>>(faces, fdata);

    raster_kernel<<<STRIPS_TOTAL / WAVES_PER_BLK, 256, 0, stream>>>(
        fdata, out_fim, out_depth, ws_w0, ws_w1);

    finalize_kernel<<<(BB * HH * WW + 255) / 256, 256, 0, stream>>>(
        textures, out_fim, ws_w0, ws_w1, out_feat);
}